// GNNModel_20925080666769
// MI455X (gfx1250) — compile-verified
//
#include <hip/hip_runtime.h>
#include <hip/hip_bf16.h>

#define N_NODES   100000
#define N_EDGES   1600000
#define NUM_GRAPHS 64
#define IN_DIM    128
#define HID       128
#define OUT_DIM   64
#define BN_EPS    1e-5f
#define M_TILES   (N_NODES / 16)   // 6250 exact
#define LDSPITCH  132              // 132 mod 64 = 4 -> conflict-free B frags

typedef float v2f __attribute__((ext_vector_type(2)));
typedef float v8f __attribute__((ext_vector_type(8)));

// ---------------------------------------------------------------------------
// f32 WMMA: D(16x16) = A(16x4) x B(4x16) + C   (V_WMMA_F32_16X16X4_F32)
// ---------------------------------------------------------------------------
__device__ __forceinline__ v8f wmma_f32x4(v2f a, v2f b, v8f c) {
  return __builtin_amdgcn_wmma_f32_16x16x4_f32(false, a, false, b, (short)0, c,
                                               false, false);
}

__device__ __forceinline__ void atomAddF(float* p, float v) {
  (void)unsafeAtomicAdd(p, v);   // -> global_atomic_add_f32
}

// ---------------------------------------------------------------------------
// zero fill (float4 granularity; all spans we pass are multiples of 4 floats)
// ---------------------------------------------------------------------------
__global__ void zero_f32(float* __restrict__ p, long n4) {
  long i = blockIdx.x * (long)blockDim.x + threadIdx.x;
  long stride = (long)gridDim.x * blockDim.x;
  float4 z = make_float4(0.f, 0.f, 0.f, 0.f);
  for (; i < n4; i += stride) ((float4*)p)[i] = z;
}

// ---------------------------------------------------------------------------
// per-graph node counts
// ---------------------------------------------------------------------------
__global__ void count_graphs(const int* __restrict__ batch, float* __restrict__ cnt) {
  long i = blockIdx.x * (long)blockDim.x + threadIdx.x;
  long stride = (long)gridDim.x * blockDim.x;
  for (; i < N_NODES; i += stride) atomAddF(&cnt[batch[i]], 1.0f);
}

// ---------------------------------------------------------------------------
// edge scatter: one wave per edge, lane handles 4 contiguous features.
// agg[dst] += feat[src]; optionally deg[dst] += 1 (first layer only).
// feat matrix (51 MB) is L2-resident on MI455X (192 MB L2) -> L2-bound.
// Prefetch next edge's source row (global_prefetch_b8) to hide gather latency
// behind the atomic-add phase.
// ---------------------------------------------------------------------------
__global__ void scatter_mean_accum(const float* __restrict__ feat,
                                   const int* __restrict__ src,
                                   const int* __restrict__ dst,
                                   float* __restrict__ agg,
                                   float* __restrict__ deg) {
  const int lane = threadIdx.x & 31;
  long wave = blockIdx.x * (long)(blockDim.x >> 5) + (threadIdx.x >> 5);
  long nw   = (long)gridDim.x * (blockDim.x >> 5);
  for (long e = wave; e < N_EDGES; e += nw) {
    int s = src[e];
    int d = dst[e];
    long en = e + nw;
    if (en < N_EDGES) {
      int sn = src[en];
      __builtin_prefetch(feat + (size_t)sn * IN_DIM + lane * 4, 0, 1);
    }
    float4 v = ((const float4*)(feat + (size_t)s * IN_DIM))[lane];
    float* ap = agg + (size_t)d * IN_DIM + lane * 4;
    atomAddF(ap + 0, v.x);
    atomAddF(ap + 1, v.y);
    atomAddF(ap + 2, v.z);
    atomAddF(ap + 3, v.w);
    if (deg != nullptr && lane == 0) atomAddF(deg + d, 1.0f);
  }
}

// ---------------------------------------------------------------------------
// One GEMM phase: acc += Arow-tile x W^T-tile over K=0..127, step 4.
// A fragment (16x4 f32): lanes 0-15 rows 0-15 K={k0,k0+1}; lanes 16-31 K={k0+2,k0+3}.
// B fragment (4x16 f32): mirror layout, B[k][n] = W[n][k], read from LDS.
// All 32 A fragments of the phase are batch-loaded up-front (fully unrolled,
// immediate offsets -> 32 in-flight global_load_b64); B fragments come from
// conflict-free ds_load_b64 (pitch 132).
// ---------------------------------------------------------------------------
template <bool SCALE>
__device__ __forceinline__ void gemm_phase(const float* __restrict__ aRow, float idg,
                                           const float* sWbase,
                                           int lh, int koff, v8f acc[8]) {
  float2 abuf[32];
#pragma unroll
  for (int k = 0; k < 32; k++)
    abuf[k] = *(const float2*)(aRow + 4 * k + koff);

  const float* bp0 = sWbase + lh * LDSPITCH + koff;
#pragma unroll
  for (int k = 0; k < 32; k++) {
    v2f a;
    a[0] = SCALE ? abuf[k].x * idg : abuf[k].x;
    a[1] = SCALE ? abuf[k].y * idg : abuf[k].y;
    float2 bv[8];
#pragma unroll
    for (int t = 0; t < 8; t++)
      bv[t] = *(const float2*)(bp0 + t * 16 * LDSPITCH + 4 * k);
#pragma unroll
    for (int t = 0; t < 8; t++) {
      v2f b;
      b[0] = bv[t].x;
      b[1] = bv[t].y;
      acc[t] = wmma_f32x4(a, b, acc[t]);
    }
  }
}

// ---------------------------------------------------------------------------
// SAGE layer GEMM: h = (agg/deg) @ wl^T + root @ wr^T + bias
// Both 128x128 weight matrices staged in LDS (pitch 132, conflict-free).
// mode 0: write h, accumulate BN sum/sumsq (regs -> LDS -> 1 global atomic/blk)
// mode 1: fuse global mean-pool: atomically add rows into pooled[batch[node]]
// One wave = 16-node row tile x all 8 column tiles (8 x v8f accumulators).
// ---------------------------------------------------------------------------
__global__ void __launch_bounds__(256)
sage_gemm(const float* __restrict__ agg,
          const float* __restrict__ deg,
          const float* __restrict__ root,
          const float* __restrict__ wl,
          const float* __restrict__ wr,
          const float* __restrict__ bias,
          float* __restrict__ hout,
          float* __restrict__ bnsum,
          float* __restrict__ bnsq,
          const int* __restrict__ batch,
          float* __restrict__ pooled,
          int mode) {
  __shared__ float sW[2 * HID * LDSPITCH];   // 132 KB: wl rows 0-127, wr rows 128-255
  __shared__ float sSum[HID];
  __shared__ float sSq[HID];

  {
    // cooperative weight staging: one row per thread (256 threads = 256 rows)
    const int r = threadIdx.x;
    const float* gsrc = (r < HID) ? (wl + (size_t)r * HID)
                                  : (wr + (size_t)(r - HID) * HID);
    float* ldst = sW + (size_t)r * LDSPITCH;   // base = r*528B, 16B aligned
#pragma unroll
    for (int c = 0; c < HID; c += 4)
      *(float4*)(ldst + c) = *(const float4*)(gsrc + c);
    if (mode == 0 && threadIdx.x < HID) {
      sSum[threadIdx.x] = 0.f;
      sSq[threadIdx.x] = 0.f;
    }
  }
  __syncthreads();

  const int lane  = threadIdx.x & 31;
  const int lh    = lane & 15;          // row (A) / col (B,D) within tile
  const int koff  = (lane >> 4) << 1;   // 0 or 2: K sub-offset for frag loads
  const int rhalf = (lane >> 4) << 3;   // 0 or 8: row offset in C/D layout

  float bias_t[8], sumAcc[8], sqAcc[8];
#pragma unroll
  for (int t = 0; t < 8; t++) {
    bias_t[t] = bias[t * 16 + lh];
    sumAcc[t] = 0.f;
    sqAcc[t]  = 0.f;
  }

  const long wave = blockIdx.x * (long)(blockDim.x >> 5) + (threadIdx.x >> 5);
  const long nw   = (long)gridDim.x * (blockDim.x >> 5);

  for (long mt = wave; mt < M_TILES; mt += nw) {
    const int row = (int)mt * 16 + lh;
    const float* aRow = agg  + (size_t)row * HID;
    const float* xRow = root + (size_t)row * HID;
    const float  idg  = 1.0f / fmaxf(deg[row], 1.0f);

    v8f acc[8];
#pragma unroll
    for (int t = 0; t < 8; t++)
#pragma unroll
      for (int r = 0; r < 8; r++) acc[t][r] = 0.f;

    gemm_phase<true >(aRow, idg, sW,                 lh, koff, acc);  // mean-agg x Wl^T
    gemm_phase<false>(xRow, 0.f, sW + HID * LDSPITCH, lh, koff, acc); // root x Wr^T

    if (mode == 0) {
#pragma unroll
      for (int t = 0; t < 8; t++) {
        const int col = t * 16 + lh;
#pragma unroll
        for (int r = 0; r < 8; r++) {
          float v = acc[t][r] + bias_t[t];
          hout[(size_t)((int)mt * 16 + rhalf + r) * HID + col] = v;
          sumAcc[t] += v;
          sqAcc[t]  += v * v;
        }
      }
    } else {
#pragma unroll
      for (int t = 0; t < 8; t++) {
        const int col = t * 16 + lh;
#pragma unroll
        for (int r = 0; r < 8; r++) {
          int node = (int)mt * 16 + rhalf + r;
          int g = batch[node];
          atomAddF(&pooled[(size_t)g * HID + col], acc[t][r] + bias_t[t]);
        }
      }
    }
  }

  if (mode == 0) {
#pragma unroll
    for (int t = 0; t < 8; t++) {
      atomicAdd(&sSum[t * 16 + lh], sumAcc[t]);   // ds_add_f32
      atomicAdd(&sSq[t * 16 + lh],  sqAcc[t]);
    }
    __syncthreads();
    if (threadIdx.x < HID)
      atomAddF(&bnsum[threadIdx.x], sSum[threadIdx.x]);
    else if (threadIdx.x < 2 * HID)
      atomAddF(&bnsq[threadIdx.x - HID], sSq[threadIdx.x - HID]);
  }
}

// ---------------------------------------------------------------------------
// BN: turn (sum, sumsq) into per-feature (scale, shift) in place
// ---------------------------------------------------------------------------
__global__ void bn_finalize(float* __restrict__ bnsum, float* __restrict__ bnsq,
                            const float* __restrict__ gamma,
                            const float* __restrict__ beta) {
  int f = threadIdx.x;
  if (f < HID) {
    float mu   = bnsum[f] * (1.0f / (float)N_NODES);
    float var  = bnsq[f]  * (1.0f / (float)N_NODES) - mu * mu;
    float rstd = rsqrtf(var + BN_EPS);
    float sc   = rstd * gamma[f];
    bnsum[f] = sc;
    bnsq[f]  = beta[f] - mu * sc;
  }
}

// ---------------------------------------------------------------------------
// elementwise BN apply + ReLU, in place, float4 vectorized
// ---------------------------------------------------------------------------
__global__ void bn_apply_relu(float* __restrict__ h,
                              const float* __restrict__ scale,
                              const float* __restrict__ shift) {
  const long n4 = (long)N_NODES * HID / 4;
  long i = blockIdx.x * (long)blockDim.x + threadIdx.x;
  long stride = (long)gridDim.x * blockDim.x;
  for (; i < n4; i += stride) {
    float4 v = ((float4*)h)[i];
    int f = (int)((i * 4) & (HID - 1));
    v.x = fmaxf(v.x * scale[f + 0] + shift[f + 0], 0.f);
    v.y = fmaxf(v.y * scale[f + 1] + shift[f + 1], 0.f);
    v.z = fmaxf(v.z * scale[f + 2] + shift[f + 2], 0.f);
    v.w = fmaxf(v.w * scale[f + 3] + shift[f + 3], 0.f);
    ((float4*)h)[i] = v;
  }
}

// ---------------------------------------------------------------------------
// final FC on pooled means: out[g][o] = (pooled[g]/cnt[g]) . fc_w[o] + fc_b[o]
// ---------------------------------------------------------------------------
__global__ void fc_out(const float* __restrict__ pooled,
                       const float* __restrict__ cnt,
                       const float* __restrict__ fcw,
                       const float* __restrict__ fcb,
                       float* __restrict__ out) {
  int tid = blockIdx.x * blockDim.x + threadIdx.x;
  if (tid >= NUM_GRAPHS * OUT_DIM) return;
  int g = tid >> 6;
  int o = tid & 63;
  float inv = 1.0f / fmaxf(cnt[g], 1.0f);
  float s = 0.f;
  const float* pr = pooled + (size_t)g * HID;
  const float* wr = fcw + (size_t)o * HID;
#pragma unroll 4
  for (int k = 0; k < HID; k++) s += pr[k] * wr[k];
  out[tid] = s * inv + fcb[o];
}

// ---------------------------------------------------------------------------
extern "C" void kernel_launch(void* const* d_in, const int* in_sizes, int n_in,
                              void* d_out, int out_size, void* d_ws, size_t ws_size,
                              hipStream_t stream) {
  const float* x     = (const float*)d_in[0];
  const int*   ei    = (const int*)d_in[1];
  const int*   batch = (const int*)d_in[2];
  const float* w1l   = (const float*)d_in[3];
  const float* b1l   = (const float*)d_in[4];
  const float* w1r   = (const float*)d_in[5];
  const float* gamma = (const float*)d_in[6];
  const float* beta  = (const float*)d_in[7];
  const float* w2l   = (const float*)d_in[8];
  const float* b2l   = (const float*)d_in[9];
  const float* w2r   = (const float*)d_in[10];
  const float* fcw   = (const float*)d_in[11];
  const float* fcb   = (const float*)d_in[12];
  float* out = (float*)d_out;

  // workspace layout (floats): zeroed span covers agg..cnt in one pass
  float* ws     = (float*)d_ws;
  float* agg    = ws;                               // N*128  (reused for layer 2)
  float* deg    = agg    + (size_t)N_NODES * HID;   // N
  float* bnsum  = deg    + N_NODES;                 // 128 (later: BN scale)
  float* bnsq   = bnsum  + HID;                     // 128 (later: BN shift)
  float* pooled = bnsq   + HID;                     // 64*128
  float* cnt    = pooled + (size_t)NUM_GRAPHS * HID;// 64
  float* h1     = cnt    + NUM_GRAPHS;              // N*128

  const int* src = ei;
  const int* dst = ei + N_EDGES;

  const long zspan = (long)N_NODES * HID + N_NODES + 2 * HID +
                     (long)NUM_GRAPHS * HID + NUM_GRAPHS;   // multiple of 4

  zero_f32<<<2048, 256, 0, stream>>>(agg, zspan / 4);
  count_graphs<<<512, 256, 0, stream>>>(batch, cnt);

  // layer 1
  scatter_mean_accum<<<2048, 256, 0, stream>>>(x, src, dst, agg, deg);
  sage_gemm<<<256, 256, 0, stream>>>(agg, deg, x, w1l, w1r, b1l,
                                     h1, bnsum, bnsq, nullptr, nullptr, 0);
  bn_finalize<<<1, 128, 0, stream>>>(bnsum, bnsq, gamma, beta);
  bn_apply_relu<<<2048, 256, 0, stream>>>(h1, bnsum, bnsq);

  // layer 2 (reuse agg; deg persists)
  zero_f32<<<2048, 256, 0, stream>>>(agg, (long)N_NODES * HID / 4);
  scatter_mean_accum<<<2048, 256, 0, stream>>>(h1, src, dst, agg, nullptr);
  sage_gemm<<<256, 256, 0, stream>>>(agg, deg, h1, w2l, w2r, b2l,
                                     nullptr, nullptr, nullptr, batch, pooled, 1);

  // readout
  fc_out<<<16, 256, 0, stream>>>(pooled, cnt, fcw, fcb, out);
}